// SparseTopKSimilarityLayer_21741124452849
// MI455X (gfx1250) — compile-verified
//
#include <hip/hip_runtime.h>
#include <hip/hip_bf16.h>

// ---------------------------------------------------------------------------
// Sparse top-k similarity search for MI455X (gfx1250, wave32).
//   queries    : (1024, 64)        f32   d_in[0]
//   table      : (4096,1,128,64)   f32   d_in[1]   (128 MB -> L2 resident)
//   prototypes : (4096, 64)        f32   d_in[2]
// Outputs (concat): values (1024,16,64) f32 | scores (1024,16) f32 | ids (1024,16) i32
// ---------------------------------------------------------------------------

typedef __attribute__((ext_vector_type(2))) float v2f;
typedef __attribute__((ext_vector_type(8))) float v8f;

#define NQ   1024
#define NC   4096
#define DIM  64
#define VPC  128      // values per cluster
#define NSR  32       // N_SEARCH
#define KTOP 16
#define NEG_INF (-3.402823e38f)

// ---------------------------------------------------------------------------
// K1: coarse GEMM  scores[q][p] = sum_d Q[q][d] * P[p][d]
// One wave per 16x16 output tile, V_WMMA_F32_16X16X4_F32, K-loop of 16 WMMAs.
// A fragment (16x4, MxK): lane<16 -> {K=k0,k0+1} of row M=lane,
//                         lane>=16 -> {K=k0+2,k0+3} of row M=lane-16.
// B fragment (4x16, KxN): same pattern with N instead of M (B = P^T, so this
// is just P rows, identical addressing to A).
// C/D layout: VGPR r, lanes 0-15: (M=r, N=lane); lanes 16-31: (M=r+8, N=lane-16)
// ---------------------------------------------------------------------------
__global__ __launch_bounds__(128) void coarse_gemm_wmma(
    const float* __restrict__ Q, const float* __restrict__ P,
    float* __restrict__ S) {
  const int wave = threadIdx.x >> 5;            // 0..3
  const int lane = threadIdx.x & 31;
  const int m0 = blockIdx.x * 16;               // 64 M tiles
  const int n0 = (blockIdx.y * 4 + wave) * 16;  // 256 N tiles
  const int half = lane >> 4;                   // 0 or 1
  const int l    = lane & 15;

  const float* qrow = Q + (m0 + l) * DIM + half * 2;
  const float* prow = P + (n0 + l) * DIM + half * 2;

  v8f c = {0.f, 0.f, 0.f, 0.f, 0.f, 0.f, 0.f, 0.f};
#pragma unroll
  for (int k0 = 0; k0 < DIM; k0 += 4) {
    v2f a = *(const v2f*)(qrow + k0);
    v2f b = *(const v2f*)(prow + k0);
    c = __builtin_amdgcn_wmma_f32_16x16x4_f32(
        /*neg_a=*/false, a, /*neg_b=*/false, b,
        /*c_mod=*/(short)0, c, /*reuse_a=*/false, /*reuse_b=*/false);
  }

  float* out = S + (size_t)(m0 + half * 8) * NC + n0 + l;
#pragma unroll
  for (int r = 0; r < 8; ++r) out[(size_t)r * NC] = c[r];
}

// ---------------------------------------------------------------------------
// K2: per-query top-32 cluster selection over 4096 scores.
// Each thread keeps its 16-score contiguous chunk in REGISTERS (all indexing
// unrolled). Per pass: unrolled local argmax -> wave32 shfl argmax (no
// barrier) -> 8-entry cross-wave merge via double-buffered LDS slots, so
// exactly ONE __syncthreads per pass (32 total). Ties -> lowest global index
// at every level (matches jax.lax.top_k ordering).
// ---------------------------------------------------------------------------
__global__ __launch_bounds__(256) void select_top32(
    const float* __restrict__ S, int* __restrict__ top_idx) {
  __shared__ float wS[2][8];
  __shared__ int   wI[2][8];

  const int tid  = threadIdx.x;
  const int lane = tid & 31;
  const int wid  = tid >> 5;
  const int q    = blockIdx.x;
  const float* s = S + (size_t)q * NC;

  // 16 contiguous scores per thread, register resident
  float c[16];
#pragma unroll
  for (int j = 0; j < 16; ++j) c[j] = s[tid * 16 + j];

  for (int pass = 0; pass < NSR; ++pass) {
    const int buf = pass & 1;
    // local argmax (keeps lowest j on tie)
    float best = c[0];
    int   bj   = 0;
#pragma unroll
    for (int j = 1; j < 16; ++j)
      if (c[j] > best) { best = c[j]; bj = j; }
    int bidx = tid * 16 + bj;
    // wave32 argmax (valid in lane 0; tie -> lowest index)
    for (int off = 16; off > 0; off >>= 1) {
      const float os = __shfl_down(best, off, 32);
      const int   oi = __shfl_down(bidx, off, 32);
      if (os > best || (os == best && oi < bidx)) { best = os; bidx = oi; }
    }
    if (lane == 0) { wS[buf][wid] = best; wI[buf][wid] = bidx; }
    __syncthreads();
    // every thread merges the 8 wave winners (broadcast LDS reads)
    float gbest = wS[buf][0];
    int   gidx  = wI[buf][0];
#pragma unroll
    for (int w = 1; w < 8; ++w) {
      const float os = wS[buf][w];
      const int   oi = wI[buf][w];
      if (os > gbest || (os == gbest && oi < gidx)) { gbest = os; gidx = oi; }
    }
    if (tid == 0) top_idx[q * NSR + pass] = gidx;
    // owner masks its register copy (predicated, unrolled -> cndmask)
    if ((gidx >> 4) == tid) {
      const int wj = gidx & 15;
#pragma unroll
      for (int j = 0; j < 16; ++j)
        if (j == wj) c[j] = NEG_INF;
    }
    // No trailing sync: next pass uses the other LDS buffer; this buffer is
    // only rewritten at pass+2, i.e. after everyone passed the sync of pass+1.
  }
}

// ---------------------------------------------------------------------------
// K3: fine stage. One wave per (query, cluster) task; 8 waves/block share one
// query (32 tasks per query, block covers 8 aligned tasks). Each lane owns 4
// consecutive values (dot-64 each, float4 loads, query broadcast from LDS),
// then wave32 shfl argmax (ties -> lowest value index, matching jnp.argmax).
// ---------------------------------------------------------------------------
__global__ __launch_bounds__(256) void fine_rows(
    const float* __restrict__ Q, const float* __restrict__ T,
    const int* __restrict__ top_idx,
    float* __restrict__ row_score, int* __restrict__ row_id) {
  __shared__ alignas(16) float qs[DIM];

  const int task0 = blockIdx.x * 8;
  const int q     = task0 >> 5;                 // same for all 8 waves
  if (threadIdx.x < DIM) qs[threadIdx.x] = Q[(size_t)q * DIM + threadIdx.x];
  __syncthreads();

  const int wid  = threadIdx.x >> 5;
  const int lane = threadIdx.x & 31;
  const int c    = (task0 + wid) & (NSR - 1);
  const int cluster = top_idx[q * NSR + c];
  const float* base = T + (size_t)cluster * VPC * DIM;

  float best = NEG_INF;
  int   bi   = 0;
#pragma unroll
  for (int vv = 0; vv < 4; ++vv) {
    const int v = lane * 4 + vv;
    const float4* vp = (const float4*)(base + (size_t)v * DIM);
    if (vv < 3) __builtin_prefetch(base + (size_t)(v + 1) * DIM, 0, 0);
    float acc = 0.f;
#pragma unroll
    for (int j = 0; j < 16; ++j) {
      const float4 x  = vp[j];
      const float4 qd = *(const float4*)(qs + j * 4);
      acc += x.x * qd.x + x.y * qd.y + x.z * qd.z + x.w * qd.w;
    }
    if (acc > best) { best = acc; bi = v; }      // tie -> lowest index
  }
  // wave32 argmax reduction (tie -> lowest index)
  for (int off = 16; off > 0; off >>= 1) {
    const float os = __shfl_down(best, off, 32);
    const int   oi = __shfl_down(bi,   off, 32);
    if (os > best || (os == best && oi < bi)) { best = os; bi = oi; }
  }
  if (lane == 0) {
    row_score[q * NSR + c] = best;
    row_id[q * NSR + c]    = bi;
  }
}

// ---------------------------------------------------------------------------
// K4: per-query top-16 of 32 row maxima via comparison ranking (one wave),
// then block-wide gather of the 16 winning 64-float value vectors.
// Rank tie-break (equal score -> higher lane first) matches the reference's
// reversed stable argsort.
// ---------------------------------------------------------------------------
__global__ __launch_bounds__(128) void final_topk_gather(
    const float* __restrict__ T, const int* __restrict__ top_idx,
    const float* __restrict__ row_score, const int* __restrict__ row_id,
    float* __restrict__ outV, float* __restrict__ outS, int* __restrict__ outI) {
  __shared__ int wCl[KTOP];
  __shared__ int wVid[KTOP];

  const int q    = blockIdx.x;
  const int wid  = threadIdx.x >> 5;
  const int lane = threadIdx.x & 31;

  if (wid == 0) {
    const float s   = row_score[q * NSR + lane];
    const int   cl  = top_idx[q * NSR + lane];
    const int   vid = row_id[q * NSR + lane];
    int rank = 0;
#pragma unroll
    for (int j = 0; j < NSR; ++j) {
      const float sj = __shfl(s, j, 32);
      if (sj > s || (sj == s && j > lane)) rank++;
    }
    if (rank < KTOP) {
      outS[q * KTOP + rank] = s;
      outI[q * KTOP + rank] = cl * VPC + vid;    // global table id
      wCl[rank]  = cl;
      wVid[rank] = vid;
    }
  }
  __syncthreads();

  for (int i = threadIdx.x; i < KTOP * DIM; i += blockDim.x) {
    const int r = i >> 6;
    const int d = i & (DIM - 1);
    outV[(size_t)q * KTOP * DIM + i] =
        T[((size_t)wCl[r] * VPC + wVid[r]) * DIM + d];
  }
}

// ---------------------------------------------------------------------------
extern "C" void kernel_launch(void* const* d_in, const int* in_sizes, int n_in,
                              void* d_out, int out_size, void* d_ws, size_t ws_size,
                              hipStream_t stream) {
  const float* Q = (const float*)d_in[0];   // 1024*64
  const float* T = (const float*)d_in[1];   // 4096*1*128*64
  const float* P = (const float*)d_in[2];   // 4096*64

  float* ws        = (float*)d_ws;
  float* scores    = ws;                                    // NQ*NC f32 (16 MB)
  int*   top_idx   = (int*)(ws + (size_t)NQ * NC);          // NQ*32 i32
  float* row_score = (float*)(top_idx + (size_t)NQ * NSR);  // NQ*32 f32
  int*   row_id    = (int*)(row_score + (size_t)NQ * NSR);  // NQ*32 i32

  float* outV = (float*)d_out;                      // NQ*16*64
  float* outS = outV + (size_t)NQ * KTOP * DIM;     // NQ*16
  int*   outI = (int*)(outS + (size_t)NQ * KTOP);   // NQ*16

  coarse_gemm_wmma<<<dim3(NQ / 16, NC / 16 / 4), 128, 0, stream>>>(Q, P, scores);
  select_top32<<<NQ, 256, 0, stream>>>(scores, top_idx);
  fine_rows<<<(NQ * NSR) / 8, 256, 0, stream>>>(Q, T, top_idx, row_score, row_id);
  final_topk_gather<<<NQ, 128, 0, stream>>>(T, top_idx, row_score, row_id, outV, outS, outI);
}